// Model_10522669875239
// MI455X (gfx1250) — compile-verified
//
#include <hip/hip_runtime.h>
#include <math.h>

#define EPSF 1e-6f

// Separating-axis test along unnormalized axis (nx,ny).
// Reference normalizes n by (||n|| + 1e-8) and uses absolute EPS; since the
// normalizer is strictly positive, multiplying the inequality through gives the
// exactly-equivalent test below with eps scaled by (||n|| + 1e-8) — no division.
__device__ __forceinline__ bool axis_separates(float nx, float ny,
                                               float x0, float y0, float x1, float y1,
                                               float ax, float ay, float bx, float by,
                                               float cx, float cy) {
    float len2 = __builtin_fmaf(nx, nx, ny * ny);
    float len  = __builtin_amdgcn_sqrtf(len2);     // single v_sqrt_f32 (1 ulp is fine under eps)
    float eps  = EPSF * (len + 1e-8f);
    // AABB projection extents: min/max over 4 corners decomposes per-coordinate.
    float px0 = nx * x0, px1 = nx * x1;
    float py0 = ny * y0, py1 = ny * y1;
    float rmin = fminf(px0, px1) + fminf(py0, py1);
    float rmax = fmaxf(px0, px1) + fmaxf(py0, py1);
    // Triangle projections.
    float pa = __builtin_fmaf(nx, ax, ny * ay);
    float pb = __builtin_fmaf(nx, bx, ny * by);
    float pc = __builtin_fmaf(nx, cx, ny * cy);
    float tmin = fminf(pa, fminf(pb, pc));
    float tmax = fmaxf(pa, fmaxf(pb, pc));
    return (rmax < tmin - eps) | (tmax < rmin - eps);   // bitwise: branchless
}

__device__ __forceinline__ float sat_hit_one(float x0, float y0, float x1, float y1,
                                             float ax, float ay, float bx, float by,
                                             float cx, float cy) {
    // Fixed axis X: robot proj = [x0, x1], triangle proj = tri x extents.
    float txmin = fminf(ax, fminf(bx, cx));
    float txmax = fmaxf(ax, fmaxf(bx, cx));
    bool sep = (x1 < txmin - EPSF) | (txmax < x0 - EPSF);
    // Fixed axis Y.
    float tymin = fminf(ay, fminf(by, cy));
    float tymax = fmaxf(ay, fmaxf(by, cy));
    sep = sep | (y1 < tymin - EPSF) | (tymax < y0 - EPSF);
    // Edge normals: e = roll(tri,-1) - tri; n = (-e.y, e.x) (unnormalized; eps scaled inside).
    sep = sep | axis_separates(ay - by, bx - ax, x0, y0, x1, y1, ax, ay, bx, by, cx, cy);
    sep = sep | axis_separates(by - cy, cx - bx, x0, y0, x1, y1, ax, ay, bx, by, cx, cy);
    sep = sep | axis_separates(cy - ay, ax - cx, x0, y0, x1, y1, ax, ay, bx, by, cx, cy);

    return sep ? 0.0f : 1.0f;
}

// Each thread handles FOUR consecutive batch elements so every global access is
// a full-width b128 op: robot_min 2x16B, robot_max 2x16B, triangles 96B = 6x16B,
// output one 16B store. 10 loads + 1 store per 176 bytes of traffic.
__global__ void __launch_bounds__(256) sat_hit_kernel(const float* __restrict__ rmin,
                                                      const float* __restrict__ rmax,
                                                      const float* __restrict__ tri,
                                                      float* __restrict__ out,
                                                      int n_quads, int n_total) {
    int i = blockIdx.x * blockDim.x + threadIdx.x;

    if (i < n_quads) {
        const float4* mnp = (const float4*)rmin + 2 * i;
        const float4* mxp = (const float4*)rmax + 2 * i;
        const float4* tp  = (const float4*)tri + 6 * i;
        float4 mn0 = mnp[0], mn1 = mnp[1];     // (x0,y0) for elements a,b | c,d
        float4 mx0 = mxp[0], mx1 = mxp[1];     // (x1,y1) for elements a,b | c,d
        float4 t0 = tp[0], t1 = tp[1], t2 = tp[2];
        float4 t3 = tp[3], t4 = tp[4], t5 = tp[5];

        float4 res;
        res.x = sat_hit_one(mn0.x, mn0.y, mx0.x, mx0.y,
                            t0.x, t0.y, t0.z, t0.w, t1.x, t1.y);
        res.y = sat_hit_one(mn0.z, mn0.w, mx0.z, mx0.w,
                            t1.z, t1.w, t2.x, t2.y, t2.z, t2.w);
        res.z = sat_hit_one(mn1.x, mn1.y, mx1.x, mx1.y,
                            t3.x, t3.y, t3.z, t3.w, t4.x, t4.y);
        res.w = sat_hit_one(mn1.z, mn1.w, mx1.z, mx1.w,
                            t4.z, t4.w, t5.x, t5.y, t5.z, t5.w);
        ((float4*)out)[i] = res;
    } else if (i == n_quads) {
        // Tail (n_total % 4 != 0) — not taken for B = 4M; kept for generality.
        for (int e = 4 * n_quads; e < n_total; ++e) {
            float2 mn = ((const float2*)rmin)[e];
            float2 mx = ((const float2*)rmax)[e];
            const float2* tp = (const float2*)tri + 3 * e;
            float2 va = tp[0], vb = tp[1], vc = tp[2];
            out[e] = sat_hit_one(mn.x, mn.y, mx.x, mx.y,
                                 va.x, va.y, vb.x, vb.y, vc.x, vc.y);
        }
    }
}

extern "C" void kernel_launch(void* const* d_in, const int* in_sizes, int n_in,
                              void* d_out, int out_size, void* d_ws, size_t ws_size,
                              hipStream_t stream) {
    (void)n_in; (void)d_ws; (void)ws_size; (void)out_size;

    const float* rmin = (const float*)d_in[0];  // (B, 2)
    const float* rmax = (const float*)d_in[1];  // (B, 2)
    const float* tri  = (const float*)d_in[2];  // (B, 3, 2)
    float* out = (float*)d_out;                 // (B, 1)

    int B = in_sizes[0] / 2;
    int n_quads = B >> 2;
    int n_threads = n_quads + ((B & 3) ? 1 : 0);

    const int block = 256;  // 8 wave32 waves per block on gfx1250
    int grid = (n_threads + block - 1) / block;
    if (grid == 0) grid = 1;

    sat_hit_kernel<<<grid, block, 0, stream>>>(rmin, rmax, tri, out, n_quads, B);
}